// HEALPixInverseRealSHT_3770981286262
// MI455X (gfx1250) — compile-verified
//
#include <hip/hip_runtime.h>
#include <hip/hip_bf16.h>

typedef __attribute__((ext_vector_type(2))) float v2f;
typedef __attribute__((ext_vector_type(8))) float v8f;

#define NSIDE_  64
#define NLAT_   255        // 4*NSIDE-1
#define LMAX_   96
#define MMAX_   96
#define BC_     256        // B*C = 4*64
#define NPIX_   49152      // 12*NSIDE^2
#define L2_     48         // LMAX/2  (k-pair count)
#define M2_     48         // MMAX/2

// ---------------- ring geometry (closed form, no host tables) ----------------
__device__ __forceinline__ int ring_nlon(int t) {
  return (t < NSIDE_ - 1) ? 4 * (t + 1)
       : (t < 3 * NSIDE_) ? 4 * NSIDE_
       : 4 * (4 * NSIDE_ - 1 - t);
}

__device__ __forceinline__ int ring_offset(int t) {
  if (t < NSIDE_ - 1) return 2 * t * (t + 1);
  if (t < 3 * NSIDE_) return 2 * (NSIDE_ - 1) * NSIDE_ + 4 * NSIDE_ * (t - (NSIDE_ - 1));
  int r = NLAT_ - t;                 // 63..1
  return NPIX_ - 2 * r * (r + 1);
}

// DFT table base (in floats) for ring size n = 4k: sum_{i<k} 96*4i = 192*k*(k-1)
__device__ __forceinline__ size_t tab_base(int k) {
  return (size_t)192 * k * (k - 1);
}
#define TAB_FLOATS 798720   // 192*64*63 + 96*256

// ------- kernel W: weighted DFT tables per distinct ring size ---------------
// cosT[base + (m2*n + j)*2 + e] =  w_m cos(2*pi*m*j/n),  m = 2*m2+e
// sinT[...]                     = -w'_m sin(2*pi*m*j/n)
// w_0=1; w_m=2 (0<m<n/2); w_{n/2}=1 cos-only; w_m=0 (m>n/2)  [irfft truncation]
__global__ void sht_build_tables(float* __restrict__ cosT, float* __restrict__ sinT) {
  const int k  = blockIdx.z + 1;             // 1..64  (n = 4k; k=64 -> 256)
  const int n  = 4 * k;
  const int m2 = blockIdx.y;                 // 0..47
  const int j  = blockIdx.x * 64 + threadIdx.x;
  if (j >= n) return;
  const int nh = n >> 1;
  const float twon = 6.28318530717958647692f / (float)n;
  float cp[2], sp[2];
  #pragma unroll
  for (int e = 0; e < 2; ++e) {
    const int mm = 2 * m2 + e;
    const float wc  = (mm == 0) ? 1.f : (mm < nh ? 2.f : (mm == nh ? 1.f : 0.f));
    const float wsn = (mm == nh) ? 0.f : wc;
    float s, c;
    __sincosf(twon * (float)((mm * j) % n), &s, &c);
    cp[e] = wc * c;
    sp[e] = -wsn * s;
  }
  const size_t o = tab_base(k) + ((size_t)m2 * n + j) * 2;
  cosT[o] = cp[0]; cosT[o + 1] = cp[1];
  sinT[o] = sp[0]; sinT[o + 1] = sp[1];
}

// ------- kernel P: repack pct[m][l][t(255)] -> pctP[m][l/2][t(256 pad)][2] ---
__global__ void sht_repack_pct(const float* __restrict__ pct,
                               float* __restrict__ pctP) {
  const int m = blockIdx.y;                  // 0..95
  const int l = blockIdx.x;                  // 0..95
  const int t = threadIdx.x;                 // 0..255
  const float v = (t < NLAT_) ? pct[((size_t)m * LMAX_ + l) * NLAT_ + t] : 0.f;
  pctP[(((size_t)m * L2_ + (l >> 1)) * 256 + t) * 2 + (l & 1)] = v;
}

// ---- kernel T: x[bc][l][m] -> xT[m][l/2][bc][2] (LDS-tiled transpose) ------
__global__ void sht_transpose_x(const float* __restrict__ xr,
                                const float* __restrict__ xi,
                                float* __restrict__ xTr,
                                float* __restrict__ xTi) {
  __shared__ float tr[32][33];
  __shared__ float ti[32][33];
  const int l   = blockIdx.z;                // 0..95
  const int m0  = blockIdx.x * 32;           // 0,32,64
  const int bc0 = blockIdx.y * 32;           // 0..224
  const int tx = threadIdx.x;                // 0..31
  const int ty = threadIdx.y;                // 0..7
  #pragma unroll
  for (int r = 0; r < 4; ++r) {
    const int bc = bc0 + ty + 8 * r;
    const int m  = m0 + tx;
    const int idx = (bc * LMAX_ + l) * MMAX_ + m;  // coalesced along m
    tr[ty + 8 * r][tx] = xr[idx];
    ti[ty + 8 * r][tx] = xi[idx];
  }
  __syncthreads();
  const size_t lbase = ((size_t)(l >> 1)) * BC_ * 2 + (l & 1);
  #pragma unroll
  for (int r = 0; r < 4; ++r) {
    const int m  = m0 + ty + 8 * r;
    const int bc = bc0 + tx;
    const size_t o = (size_t)m * L2_ * BC_ * 2 + lbase + (size_t)bc * 2;
    xTr[o] = tr[tx][ty + 8 * r];
    xTi[o] = ti[tx][ty + 8 * r];
  }
}

// ------- kernel A: Legendre GEMM (WMMA f32 16x16x4) + phase rotation --------
// For fixed m:  D[t,bc] = sum_l pct[m][l][t] * xT[m][l][bc]
// 2x register blocking along bc: one pct A-fragment feeds 4 WMMAs.
// Epilogue: a = re*cos(th) - im*sin(th); b = re*sin(th) + im*cos(th),
// th = offset[t][m]; store into coef[t][m/2][bc][2].
__global__ __launch_bounds__(32)
void sht_legendre_rotate(const float* __restrict__ xTr,     // [m][l2][bc][2]
                         const float* __restrict__ xTi,
                         const float* __restrict__ pctP,    // [m][l2][t256][2]
                         const float* __restrict__ offset,  // [t][m]
                         float* __restrict__ cA,            // [t][m2][bc][2]
                         float* __restrict__ cB) {
  const int lane   = threadIdx.x;
  const int bcpair = blockIdx.x;     // 0..7  (two 16-wide bc tiles each)
  const int ttile  = blockIdx.y;     // 0..15 (M tiles, t; row 255 is pad)
  const int m      = blockIdx.z;     // 0..95
  const int hi  = lane >> 4;         // 0/1
  const int lid = lane & 15;
  const int trow = ttile * 16 + lid;             // A row (t), pad row is zeros
  const int bc0  = bcpair * 32 + lid;            // B col, tile 0 (tile 1 = +16)

  const float* __restrict__ pb = pctP + ((size_t)m * L2_ * 256 + trow) * 2;
  const float* __restrict__ xr = xTr + ((size_t)m * L2_ * BC_ + bc0) * 2;
  const float* __restrict__ xi = xTi + ((size_t)m * L2_ * BC_ + bc0) * 2;

  v8f accR0 = {0.f, 0.f, 0.f, 0.f, 0.f, 0.f, 0.f, 0.f};
  v8f accI0 = accR0, accR1 = accR0, accI1 = accR0;

  for (int kk = 0; kk < LMAX_; kk += 4) {
    const int l2 = (kk >> 1) + hi;               // lane's k-pair index
    const v2f af  = *(const v2f*)(pb + (size_t)l2 * 256 * 2);  // b64, coalesced
    const v2f bR0 = *(const v2f*)(xr + (size_t)l2 * BC_ * 2);
    const v2f bR1 = *(const v2f*)(xr + (size_t)l2 * BC_ * 2 + 32);
    const v2f bI0 = *(const v2f*)(xi + (size_t)l2 * BC_ * 2);
    const v2f bI1 = *(const v2f*)(xi + (size_t)l2 * BC_ * 2 + 32);
    accR0 = __builtin_amdgcn_wmma_f32_16x16x4_f32(false, af, false, bR0,
                                                  (short)0, accR0, false, false);
    accI0 = __builtin_amdgcn_wmma_f32_16x16x4_f32(false, af, false, bI0,
                                                  (short)0, accI0, false, false);
    accR1 = __builtin_amdgcn_wmma_f32_16x16x4_f32(false, af, false, bR1,
                                                  (short)0, accR1, false, false);
    accI1 = __builtin_amdgcn_wmma_f32_16x16x4_f32(false, af, false, bI1,
                                                  (short)0, accI1, false, false);
  }

  // D layout: col bc = lid, row t = v + 8*hi (+ ttile*16)
  #pragma unroll
  for (int v = 0; v < 8; ++v) {
    const int t = ttile * 16 + v + 8 * hi;
    if (t < NLAT_) {
      float s, c;
      __sincosf(offset[t * MMAX_ + m], &s, &c);
      const size_t o = (((size_t)t * M2_ + (m >> 1)) * BC_ + bc0) * 2 + (m & 1);
      float re = accR0[v], im = accI0[v];
      cA[o]      = re * c - im * s;              // Re(X) after rotation
      cB[o]      = re * s + im * c;              // Im(X) after rotation
      re = accR1[v]; im = accI1[v];
      cA[o + 32] = re * c - im * s;
      cB[o + 32] = re * s + im * c;
    }
  }
}

// -------- kernel B: per-ring inverse rDFT as GEMM (WMMA f32 16x16x4) --------
// y[bc, j] = sum_m a[bc,m]*cosT[n][m][j] + b[bc,m]*sinT[n][m][j]
// 2x register blocking along bc: table fragments feed 4 WMMAs.
__global__ __launch_bounds__(32)
void sht_ring_idft(const float* __restrict__ cA,    // [t][m2][bc][2]
                   const float* __restrict__ cB,
                   const float* __restrict__ cosT,  // per-size [m2][j][2]
                   const float* __restrict__ sinT,
                   float* __restrict__ out) {       // [bc][NPIX]
  const int lane   = threadIdx.x;
  const int jtile  = blockIdx.x;   // 0..15
  const int bcpair = blockIdx.y;   // 0..7 (two 16-wide bc tiles each)
  const int t      = blockIdx.z;   // 0..254
  const int n = ring_nlon(t);
  if (jtile * 16 >= n) return;     // uniform per block: EXEC stays full
  const int roff = ring_offset(t);
  const int hi  = lane >> 4;
  const int lid = lane & 15;
  const int bc0 = bcpair * 32 + lid;     // A row, tile 0 (tile 1 = +16)
  const int j   = jtile * 16 + lid;      // B col / D col (lon index)

  const float* __restrict__ aB = cA + ((size_t)t * M2_ * BC_ + bc0) * 2;
  const float* __restrict__ bB = cB + ((size_t)t * M2_ * BC_ + bc0) * 2;
  const size_t tb = tab_base(n >> 2) + (size_t)j * 2;
  const float* __restrict__ cT = cosT + tb;
  const float* __restrict__ sT = sinT + tb;

  v8f acc0 = {0.f, 0.f, 0.f, 0.f, 0.f, 0.f, 0.f, 0.f};
  v8f acc1 = acc0;

  for (int kk = 0; kk < MMAX_; kk += 4) {
    const int m2 = (kk >> 1) + hi;               // lane's k-pair index
    const v2f cf  = *(const v2f*)(cT + (size_t)m2 * n * 2);    // b64, coalesced
    const v2f sf  = *(const v2f*)(sT + (size_t)m2 * n * 2);
    const v2f af0 = *(const v2f*)(aB + (size_t)m2 * BC_ * 2);
    const v2f af1 = *(const v2f*)(aB + (size_t)m2 * BC_ * 2 + 32);
    const v2f bf0 = *(const v2f*)(bB + (size_t)m2 * BC_ * 2);
    const v2f bf1 = *(const v2f*)(bB + (size_t)m2 * BC_ * 2 + 32);
    acc0 = __builtin_amdgcn_wmma_f32_16x16x4_f32(false, af0, false, cf,
                                                 (short)0, acc0, false, false);
    acc0 = __builtin_amdgcn_wmma_f32_16x16x4_f32(false, bf0, false, sf,
                                                 (short)0, acc0, false, false);
    acc1 = __builtin_amdgcn_wmma_f32_16x16x4_f32(false, af1, false, cf,
                                                 (short)0, acc1, false, false);
    acc1 = __builtin_amdgcn_wmma_f32_16x16x4_f32(false, bf1, false, sf,
                                                 (short)0, acc1, false, false);
  }

  // D layout: col j = lid (+jtile*16), row bc = v + 8*hi (+ bc tile base)
  #pragma unroll
  for (int v = 0; v < 8; ++v) {
    const int bc = bcpair * 32 + v + 8 * hi;
    if (j < n) {
      out[(size_t)bc * NPIX_ + roff + j]        = acc0[v];  // coalesced (j)
      out[(size_t)(bc + 16) * NPIX_ + roff + j] = acc1[v];
    }
  }
}

// ------------------------------- launch -------------------------------------
extern "C" void kernel_launch(void* const* d_in, const int* in_sizes, int n_in,
                              void* d_out, int out_size, void* d_ws, size_t ws_size,
                              hipStream_t stream) {
  (void)in_sizes; (void)n_in; (void)out_size; (void)ws_size;
  const float* x_real = (const float*)d_in[0];   // (4,64,96,96)
  const float* x_imag = (const float*)d_in[1];   // (4,64,96,96)
  const float* pct    = (const float*)d_in[2];   // (96,96,255)
  const float* offset = (const float*)d_in[3];   // (255,96)
  float* out = (float*)d_out;                    // (4,64,49152)

  // workspace (floats): xT_re, xT_im, pctP, cA, cB, cosT, sinT  (~85 MB)
  float* xTr  = (float*)d_ws;
  float* xTi  = xTr  + (size_t)MMAX_ * L2_ * BC_ * 2;
  float* pctP = xTi  + (size_t)MMAX_ * L2_ * BC_ * 2;
  float* cA   = pctP + (size_t)MMAX_ * L2_ * 256 * 2;
  float* cB   = cA   + (size_t)NLAT_ * M2_ * BC_ * 2;
  float* cosT = cB   + (size_t)NLAT_ * M2_ * BC_ * 2;
  float* sinT = cosT + (size_t)TAB_FLOATS;

  sht_build_tables<<<dim3(4, M2_, 64), dim3(64), 0, stream>>>(cosT, sinT);

  sht_repack_pct<<<dim3(LMAX_, MMAX_), dim3(256), 0, stream>>>(pct, pctP);

  sht_transpose_x<<<dim3(3, 8, LMAX_), dim3(32, 8), 0, stream>>>(
      x_real, x_imag, xTr, xTi);

  sht_legendre_rotate<<<dim3(8, 16, MMAX_), dim3(32), 0, stream>>>(
      xTr, xTi, pctP, offset, cA, cB);

  sht_ring_idft<<<dim3(16, 8, NLAT_), dim3(32), 0, stream>>>(
      cA, cB, cosT, sinT, out);
}